// PatchEmbed_HS_MAE_13073880449623
// MI455X (gfx1250) — compile-verified
//
#include <hip/hip_runtime.h>

typedef __attribute__((ext_vector_type(2))) float v2f;
typedef __attribute__((ext_vector_type(8))) float v8f;

#define B_SZ   512
#define C_SZ   3
#define H_SZ   224
#define W_SZ   224
#define NPATCH 1024
#define EMBED  768
#define HW     (H_SZ * W_SZ)

// One wave owns a 16-row (b,n) tile, sweeps e in chunks of 32 (two 16x16 WMMA
// tiles per iteration so each output row gets a full 128B line of NT stores).
// A (16x4 fp32, K=3 zero-padded) gathered once per wave, fully branchless.
__global__ __launch_bounds__(256) void patch_embed_wmma_f32(
    const float* __restrict__ x,       // [B, C, H, W]
    const int*   __restrict__ curve,   // [NPATCH, 2]  (x, y)
    const float* __restrict__ proj_w,  // [EMBED, C]
    const float* __restrict__ proj_b,  // [EMBED]
    float*       __restrict__ out)     // [B, NPATCH, EMBED]
{
    const int lane  = threadIdx.x & 31;
    const int waveb = threadIdx.x >> 5;
    const int half  = lane >> 4;       // 0: lanes 0-15 (K=0,1), 1: lanes 16-31 (K=2,3)
    const int l16   = lane & 15;

    const long row_tile = (long)blockIdx.x * 8 + waveb;   // 16-row tile index
    // M = B*NPATCH = 524288 = 32768 tiles exactly; grid sized to match.

    // ---- Gather A tile (branchless): row m = row_tile*16 + l16 ----
    const long m  = row_tile * 16 + l16;
    const int  b  = (int)(m >> 10);        // m / NPATCH
    const int  n  = (int)(m & 1023);       // m % NPATCH
    const int  cx = curve[2 * n + 0];
    const int  cy = curve[2 * n + 1];
    const long base = (long)b * C_SZ * HW + (long)cy * W_SZ + cx;

    // half==0 lanes carry K={0,1} (ch 0,1); half==1 lanes carry K={2,3} (ch 2, pad)
    const float ax = x[base + (long)(half * 2) * HW];  // ch0 or ch2
    const float ay = x[base + HW];                     // ch1 (ignored by half==1)
    v2f a;
    a.x = ax;
    a.y = half ? 0.0f : ay;

    const long row0 = row_tile * 16 + (long)half * 8;  // first M row this half stores

    for (int it = 0; it < EMBED / 32; ++it) {
        const int e0 = it * 32 + l16;   // column in low tile
        const int e1 = e0 + 16;         // column in high tile

        // ---- B tiles 4x16 (branchless): column e, rows K mirror A's half split
        v2f blo, bhi;
        blo.x = proj_w[e0 * 3 + half * 2];
        bhi.x = proj_w[e1 * 3 + half * 2];
        const float blo1 = proj_w[e0 * 3 + 1];
        const float bhi1 = proj_w[e1 * 3 + 1];
        blo.y = half ? 0.0f : blo1;
        bhi.y = half ? 0.0f : bhi1;

        // ---- C = bias broadcast (each accum VGPR = distinct M row, same column)
        const float bias0 = proj_b[e0];
        const float bias1 = proj_b[e1];
        v8f c0, c1;
        #pragma unroll
        for (int r = 0; r < 8; ++r) { c0[r] = bias0; c1[r] = bias1; }

        v8f d0 = __builtin_amdgcn_wmma_f32_16x16x4_f32(
            false, a, false, blo, (short)0, c0, false, false);
        v8f d1 = __builtin_amdgcn_wmma_f32_16x16x4_f32(
            false, a, false, bhi, (short)0, c1, false, false);

        // ---- Store: lanes 0-15 of d0 + d1 cover one full 128B line per row ----
        float* o = out + row0 * EMBED + e0;
        #pragma unroll
        for (int r = 0; r < 8; ++r) {
            __builtin_nontemporal_store(d0[r], o + (long)r * EMBED);
            __builtin_nontemporal_store(d1[r], o + (long)r * EMBED + 16);
        }
    }
}

extern "C" void kernel_launch(void* const* d_in, const int* in_sizes, int n_in,
                              void* d_out, int out_size, void* d_ws, size_t ws_size,
                              hipStream_t stream) {
    (void)in_sizes; (void)n_in; (void)d_ws; (void)ws_size; (void)out_size;
    const float* x      = (const float*)d_in[0];
    const int*   curve  = (const int*)  d_in[1];
    const float* proj_w = (const float*)d_in[2];
    const float* proj_b = (const float*)d_in[3];
    float*       out    = (float*)d_out;

    // M = 512*1024 = 524288 rows -> 32768 16-row tiles -> 8 waves/block -> 4096 blocks
    const int blocks = (B_SZ * NPATCH) / 16 / 8;
    patch_embed_wmma_f32<<<blocks, 256, 0, stream>>>(x, curve, proj_w, proj_b, out);
}